// EncoderLayer_80753975099983
// MI455X (gfx1250) — compile-verified
//
#include <hip/hip_runtime.h>
#include <math.h>

// ---------------------------------------------------------------------------
// HGT-style sparse attention layer for MI455X (gfx1250, wave32).
//   qkv = x @ W + b        -> fp32 WMMA 16x16x4 (exact, matches fp32 ref)
//   per-edge softmax-attn  -> 3 edge passes (qk+segmax, exp+segsum, scatter)
// Gathers are L2-resident (qkv buffer 123MB < 192MB L2); HBM floor ~20us.
// ---------------------------------------------------------------------------

#define NN 20000
#define HH 4
#define DD 128
#define EE 320000
#define QKVC (3 * DD)                 // 384
#define SEGN (NN * HH)                // 80000

__device__ __constant__ float kTemp = 0.08838834764831845f;  // 1/sqrt(128)

typedef __attribute__((ext_vector_type(2))) float v2f;
typedef __attribute__((ext_vector_type(8))) float v8f;

// ---- helpers: monotone uint encoding of fp32 for atomic segment-max -------
__device__ __forceinline__ unsigned enc_f32(float f) {
    unsigned u = __float_as_uint(f);
    return (u & 0x80000000u) ? ~u : (u | 0x80000000u);
}
__device__ __forceinline__ float dec_f32(unsigned e) {
    return (e & 0x80000000u) ? __uint_as_float(e & 0x7FFFFFFFu)
                             : __uint_as_float(~e);
}

// ---- init kernels ---------------------------------------------------------
__global__ void zero_f32_kernel(float* __restrict__ p, int n) {
    int i = blockIdx.x * blockDim.x + threadIdx.x;
    if (i < n) p[i] = 0.0f;
}

__global__ void seg_init_kernel(unsigned* __restrict__ menc,
                                float* __restrict__ den, int n) {
    int i = blockIdx.x * blockDim.x + threadIdx.x;
    if (i < n) { menc[i] = 0u; den[i] = 0.0f; }  // enc(-NaN)=0: identity for max
}

// ---- QKV projection GEMM via V_WMMA_F32_16X16X4_F32 -----------------------
// A = X (N*H, 128) row-major, B = W (128, 384) row-major, OUT = (N*H, 384).
// One wave per 16x16 output tile; K loop of 32 WMMAs.
// A 16x4 layout: lane l holds A[m=l&15][koff..koff+1], koff = 2*(l>>4).
// B 4x16 layout: lane l holds B[koff..koff+1][n=l&15].
// C/D: VGPR r -> row r (lanes 0-15) and row r+8 (lanes 16-31), col = lane&15.
__global__ __launch_bounds__(128) void qkv_gemm_kernel(
    const float* __restrict__ X, const float* __restrict__ W,
    const float* __restrict__ BIAS, float* __restrict__ OUT) {
    const int lane    = threadIdx.x & 31;
    const int wave    = threadIdx.x >> 5;
    const int rowBase = blockIdx.x * 16;                 // 5000 row tiles
    const int colBase = (blockIdx.y * 4 + wave) * 16;    // 24 col tiles / 4 waves
    const int m       = lane & 15;                       // row (A) / col (B)
    const int half    = lane >> 4;
    const int koff    = half * 2;

    const float* arow = X + (size_t)(rowBase + m) * DD;
    const float* bcol = W + colBase + m;

    v8f acc = {};
#pragma unroll 4
    for (int k0 = 0; k0 < DD; k0 += 4) {
        v2f a;
        a[0] = arow[k0 + koff];
        a[1] = arow[k0 + koff + 1];
        v2f b;
        b[0] = bcol[(size_t)(k0 + koff) * QKVC];
        b[1] = bcol[(size_t)(k0 + koff + 1) * QKVC];
        acc = __builtin_amdgcn_wmma_f32_16x16x4_f32(
            /*neg_a=*/false, a, /*neg_b=*/false, b,
            /*c_mod=*/(short)0, acc, /*reuse_a=*/false, /*reuse_b=*/false);
    }

    const float bias = BIAS[colBase + m];
    float* ocol = OUT + colBase + m;
#pragma unroll
    for (int r = 0; r < 8; ++r)
        ocol[(size_t)(rowBase + r + 8 * half) * QKVC] = acc[r] + bias;
}

// ---- pass 1: per-edge qk dot + atomic segment max -------------------------
// grid = E, block = 128 (4 waves; wave w = head w). Lane owns 4 channels.
__global__ __launch_bounds__(128) void edge_qk_kernel(
    const float* __restrict__ QKV, const int* __restrict__ SRC,
    const int* __restrict__ DST, float* __restrict__ QK,
    unsigned* __restrict__ MENC) {
    const int e = blockIdx.x;
    const int h = threadIdx.x >> 5;
    const int lane = threadIdx.x & 31;
    const int s = SRC[e], d = DST[e];
    const float4 q = *(const float4*)(QKV + ((size_t)s * HH + h) * QKVC + lane * 4);
    const float4 k = *(const float4*)(QKV + ((size_t)d * HH + h) * QKVC + DD + lane * 4);
    float dot = q.x * k.x + q.y * k.y + q.z * k.z + q.w * k.w;
#pragma unroll
    for (int off = 16; off > 0; off >>= 1) dot += __shfl_xor(dot, off, 32);
    if (lane == 0) {
        dot *= kTemp;
        QK[(size_t)e * HH + h] = dot;
        atomicMax(MENC + (size_t)s * HH + h, enc_f32(dot));
    }
}

// ---- pass 2: ex = exp(qk - m[src]); atomic segment sum --------------------
__global__ void edge_exp_kernel(const int* __restrict__ SRC,
                                float* __restrict__ QK,
                                const unsigned* __restrict__ MENC,
                                float* __restrict__ DEN) {
    int i = blockIdx.x * blockDim.x + threadIdx.x;  // over E*H
    if (i >= EE * HH) return;
    const int e = i >> 2, h = i & 3;
    const int s = SRC[e];
    const float mx = dec_f32(MENC[(size_t)s * HH + h]);
    const float ex = __expf(QK[i] - mx);
    QK[i] = ex;
    atomicAdd(DEN + (size_t)s * HH + h, ex);
}

// ---- pass 3: out[src] += pw * (ex/den[src]) * v[dst] ----------------------
__global__ __launch_bounds__(128) void edge_av_kernel(
    const float* __restrict__ QKV, const int* __restrict__ SRC,
    const int* __restrict__ DST, const float* __restrict__ EX,
    const float* __restrict__ DEN, const float* __restrict__ PW, int pwIdx,
    float* __restrict__ OUT) {
    const int e = blockIdx.x;
    const int h = threadIdx.x >> 5;
    const int lane = threadIdx.x & 31;
    const int s = SRC[e], d = DST[e];
    const float alpha =
        PW[pwIdx] * EX[(size_t)e * HH + h] / DEN[(size_t)s * HH + h];
    const float4 v =
        *(const float4*)(QKV + ((size_t)d * HH + h) * QKVC + 2 * DD + lane * 4);
    float* o = OUT + ((size_t)s * HH + h) * DD + lane * 4;
    atomicAdd(o + 0, alpha * v.x);
    atomicAdd(o + 1, alpha * v.y);
    atomicAdd(o + 2, alpha * v.z);
    atomicAdd(o + 3, alpha * v.w);
}

// ---- per-branch driver ----------------------------------------------------
static void run_branch(const float* x, const float* W, const float* bias,
                       const float* pw, const int* adj0, const int* adj1,
                       float* out, float* qkv, float* qk, unsigned* menc,
                       float* den, hipStream_t stream) {
    dim3 gGemm(SEGN / 16, QKVC / 64);  // 5000 x 6, 4 waves/block
    qkv_gemm_kernel<<<gGemm, 128, 0, stream>>>(x, W, bias, qkv);
    const int* adjs[2] = {adj0, adj1};
    for (int i = 0; i < 2; ++i) {
        const int* src = adjs[i];
        const int* dst = adjs[i] + EE;  // adj is (2, E): row0 = src, row1 = dst
        seg_init_kernel<<<(SEGN + 255) / 256, 256, 0, stream>>>(menc, den, SEGN);
        edge_qk_kernel<<<EE, 128, 0, stream>>>(qkv, src, dst, qk, menc);
        edge_exp_kernel<<<(EE * HH + 255) / 256, 256, 0, stream>>>(src, qk, menc, den);
        edge_av_kernel<<<EE, 128, 0, stream>>>(qkv, src, dst, qk, den, pw, i, out);
    }
}

extern "C" void kernel_launch(void* const* d_in, const int* in_sizes, int n_in,
                              void* d_out, int out_size, void* d_ws,
                              size_t ws_size, hipStream_t stream) {
    const float* v      = (const float*)d_in[0];
    const float* c      = (const float*)d_in[1];
    const float* pw_var = (const float*)d_in[2];
    const float* pw_cls = (const float*)d_in[3];
    const float* W_var  = (const float*)d_in[4];
    const float* b_var  = (const float*)d_in[5];
    const float* W_cls  = (const float*)d_in[6];
    const float* b_cls  = (const float*)d_in[7];
    const int* mp_var_0 = (const int*)d_in[8];
    const int* mp_var_1 = (const int*)d_in[9];
    const int* mp_cls_0 = (const int*)d_in[10];
    const int* mp_cls_1 = (const int*)d_in[11];

    float* out_v = (float*)d_out;                    // (N,H,D)
    float* out_c = out_v + (size_t)NN * HH * DD;     // (N,H,D)

    // Workspace layout (reused across branches): ~128.6 MB total.
    float* qkv     = (float*)d_ws;                          // 30.72M f32
    float* qk      = qkv + (size_t)SEGN * QKVC;             // 1.28M f32 (qk->ex)
    unsigned* menc = (unsigned*)(qk + (size_t)EE * HH);     // 80000 u32
    float* den     = (float*)(menc + SEGN);                 // 80000 f32

    const int totOut = 2 * SEGN * DD;
    zero_f32_kernel<<<(totOut + 255) / 256, 256, 0, stream>>>(out_v, totOut);

    run_branch(v, W_var, b_var, pw_var, mp_var_0, mp_var_1, out_v,
               qkv, qk, menc, den, stream);
    run_branch(c, W_cls, b_cls, pw_cls, mp_cls_0, mp_cls_1, out_c,
               qkv, qk, menc, den, stream);
}